// FisheyeConv_11931419148438
// MI455X (gfx1250) — compile-verified
//
#include <hip/hip_runtime.h>
#include <math.h>

typedef __attribute__((ext_vector_type(2))) float v2f;
typedef __attribute__((ext_vector_type(8))) float v8f;

#define H_    48
#define W_    48
#define NPIX  2304
#define CIN   32
#define COUT  64
#define LL    4
#define KNB   29
#define KPAD  32
#define NL    9216
#define RSTRIDE 12           // R rows padded 9 -> 12 floats for float4 loads
#define EPS_  1e-7f
#define PI_   3.14159265358979323846f
#define LOG2E_ 1.44269504088896341f

__device__ __forceinline__ v8f wmma4(v2f a, v2f b, v8f c) {
  // D = A(16x4,f32) * B(4x16,f32) + C(16x16,f32)
  return __builtin_amdgcn_wmma_f32_16x16x4_f32(false, a, false, b, (short)0, c, false, false);
}

// swish(x) = x * sigmoid(x) = x / (1 + exp(-x)), via v_exp_f32 + v_rcp_f32.
// x -> -inf: exp2(+big)=inf, rcp=0, result 0.  x -> +inf: rcp(1)=1, result x.
__device__ __forceinline__ float swishf(float x) {
  float e = __builtin_amdgcn_exp2f(-LOG2E_ * x);
  return x * __builtin_amdgcn_rcpf(1.f + e);
}

__device__ __forceinline__ void so3_exp3(float wx, float wy, float wz, float R[9]) {
  float th = sqrtf(wx * wx + wy * wy + wz * wz);
  th = fmaxf(th, EPS_);
  float kx = wx / th, ky = wy / th, kz = wz / th;
  float s = sinf(th), c = 1.f - cosf(th);
  R[0] = 1.f - c * (ky * ky + kz * kz);
  R[1] = -s * kz + c * (kx * ky);
  R[2] = s * ky + c * (kx * kz);
  R[3] = s * kz + c * (kx * ky);
  R[4] = 1.f - c * (kx * kx + kz * kz);
  R[5] = -s * kx + c * (ky * kz);
  R[6] = -s * ky + c * (kx * kz);
  R[7] = s * kx + c * (ky * kz);
  R[8] = 1.f - c * (kx * kx + ky * ky);
}

// ---------------------------------------------------------------------------
// Kernel 1: Rlift[i] = Rp(pixel) @ Rz(l) for i = p*4 + l, padded rows of 12.
// ---------------------------------------------------------------------------
__global__ void k_build_R(float* __restrict__ Rbuf) {
  int i = blockIdx.x * blockDim.x + threadIdx.x;
  if (i >= NL) return;
  int p = i >> 2, l = i & 3;
  float px = (float)(p % W_), py = (float)(p / W_);
  float dx = px - 23.5f, dy = py - 23.5f;
  float r = sqrtf(dx * dx + dy * dy);
  float theta = r * (PI_ / 48.0f);             // r / FOCAL, FOCAL = W/pi
  float rs = fmaxf(r, EPS_);
  float st = sinf(theta);
  float rx = st * dx / rs, ry = st * dy / rs, rz = cosf(theta);
  float pz = fminf(fmaxf(rz, -1.f + 1e-6f), 1.f - 1e-6f);
  float ang = acosf(pz);
  float ax = -ry, ay = rx;
  float an = fmaxf(sqrtf(ax * ax + ay * ay), EPS_);
  ax /= an; ay /= an;
  float Rp[9];
  so3_exp3(ax * ang, ay * ang, 0.f, Rp);
  float alpha = (float)l * (2.f * PI_ / (float)LL);
  float ca = cosf(alpha), sa = sinf(alpha);
  float Ro[9];
#pragma unroll
  for (int rr = 0; rr < 3; rr++) {
    Ro[rr * 3 + 0] = Rp[rr * 3 + 0] * ca + Rp[rr * 3 + 1] * sa;
    Ro[rr * 3 + 1] = -Rp[rr * 3 + 0] * sa + Rp[rr * 3 + 1] * ca;
    Ro[rr * 3 + 2] = Rp[rr * 3 + 2];
  }
  float4* o4 = (float4*)(Rbuf + i * RSTRIDE);  // 48B row, 16B aligned
  o4[0] = make_float4(Ro[0], Ro[1], Ro[2], Ro[3]);
  o4[1] = make_float4(Ro[4], Ro[5], Ro[6], Ro[7]);
  o4[2] = make_float4(Ro[8], 0.f, 0.f, 0.f);
}

// ---------------------------------------------------------------------------
// Kernel 2: one wave per pixel p. For each neighbor j (29), push the 16
// (l,l2) rows through the 3->32->32->16 MLP with f32 WMMA, then
// wagg[p][j][f] = sum over the 16 rows.  Padded to j<32 with zeros.
// ---------------------------------------------------------------------------
__global__ void __launch_bounds__(128)
k_wagg(const float* __restrict__ Rbuf, const int* __restrict__ nbidx,
       const float* __restrict__ W1, const float* __restrict__ b1,
       const float* __restrict__ W2, const float* __restrict__ b2,
       const float* __restrict__ W3, const float* __restrict__ b3,
       float* __restrict__ wagg) {
  __shared__ float hbuf[4][16 * 32];            // per-wave activation stage
  int tid = threadIdx.x;
  int wv = tid >> 5;
  int lane = tid & 31;
  int row = lane & 15;                          // MLP row (l*4+l2); dup in high half
  int half = lane >> 4;                         // lane half selects K pair / M half
  int col = row;                                // N index for B / C fragments
  int p = blockIdx.x * 4 + wv;
  int l = row >> 2, l2 = row & 3;

  // --- B fragments (weights), f32 WMMA layout: VGPRj holds K = 4kc+j+2*half
  v2f w1f[2], w2f[8][2], w3f[8];
#pragma unroll
  for (int t = 0; t < 2; t++) {
    int c = t * 16 + col;
    w1f[t].x = W1[(2 * half) * 32 + c];          // K=0 or 2
    w1f[t].y = half ? 0.f : W1[1 * 32 + c];      // K=1 or 3 (pad)
  }
#pragma unroll
  for (int kc = 0; kc < 8; kc++) {
#pragma unroll
    for (int t = 0; t < 2; t++) {
      int c = t * 16 + col;
      w2f[kc][t].x = W2[(4 * kc + 2 * half) * 32 + c];
      w2f[kc][t].y = W2[(4 * kc + 1 + 2 * half) * 32 + c];
    }
    w3f[kc].x = W3[(4 * kc + 2 * half) * 16 + col];
    w3f[kc].y = W3[(4 * kc + 1 + 2 * half) * 16 + col];
  }
  float b1v0 = b1[col], b1v1 = b1[16 + col];
  float b2v0 = b2[col], b2v1 = b2[16 + col];
  float b3v = b3[col];

  float Rq[9];
  {
    const float4* q4 = (const float4*)(Rbuf + (p * 4 + l) * RSTRIDE);
    float4 q0 = q4[0], q1 = q4[1], q2 = q4[2];
    Rq[0] = q0.x; Rq[1] = q0.y; Rq[2] = q0.z; Rq[3] = q0.w;
    Rq[4] = q1.x; Rq[5] = q1.y; Rq[6] = q1.z; Rq[7] = q1.w;
    Rq[8] = q2.x;
  }

  for (int j = 0; j < KNB; j++) {
    int idx = nbidx[p * KNB + j];
    float validf = (idx >= 0) ? 1.f : 0.f;
    int pix = idx < 0 ? 0 : idx;
    float Rb[9];
    {
      const float4* b4 = (const float4*)(Rbuf + (pix * 4 + l2) * RSTRIDE);
      float4 q0 = b4[0], q1 = b4[1], q2 = b4[2];
      Rb[0] = q0.x; Rb[1] = q0.y; Rb[2] = q0.z; Rb[3] = q0.w;
      Rb[4] = q1.x; Rb[5] = q1.y; Rb[6] = q1.z; Rb[7] = q1.w;
      Rb[8] = q2.x;
    }
    // Rab = Rb^T @ Rq ;  A(i,k) = sum_j Rb[j][i]*Rq[j][k]
    float A01 = Rb[0] * Rq[1] + Rb[3] * Rq[4] + Rb[6] * Rq[7];
    float A02 = Rb[0] * Rq[2] + Rb[3] * Rq[5] + Rb[6] * Rq[8];
    float A10 = Rb[1] * Rq[0] + Rb[4] * Rq[3] + Rb[7] * Rq[6];
    float A12 = Rb[1] * Rq[2] + Rb[4] * Rq[5] + Rb[7] * Rq[8];
    float A20 = Rb[2] * Rq[0] + Rb[5] * Rq[3] + Rb[8] * Rq[6];
    float A21 = Rb[2] * Rq[1] + Rb[5] * Rq[4] + Rb[8] * Rq[7];
    float A00 = Rb[0] * Rq[0] + Rb[3] * Rq[3] + Rb[6] * Rq[6];
    float A11 = Rb[1] * Rq[1] + Rb[4] * Rq[4] + Rb[7] * Rq[7];
    float A22 = Rb[2] * Rq[2] + Rb[5] * Rq[5] + Rb[8] * Rq[8];
    float tr = A00 + A11 + A22;
    float cth = fminf(fmaxf((tr - 1.f) * 0.5f, -1.f + 1e-6f), 1.f - 1e-6f);
    float th = acosf(cth);
    // th / (2 sin th) with sin(acos(c)) = sqrt(1-c^2); 1-c^2 >= ~2e-6
    float sc = 0.5f * th * __builtin_amdgcn_rsqf(1.f - cth * cth);
    float abx = (A21 - A12) * sc;
    float aby = (A02 - A20) * sc;
    float abz = (A10 - A01) * sc;

    // ---- layer 1: 16x3 @ 3x32 (K padded to 4) ----
    v2f a1;
    a1.x = half ? abz : abx;                      // K=0 / K=2
    a1.y = half ? 0.f : aby;                      // K=1 / K=3(pad)
    v8f c0 = {}, c1 = {};
    c0 = wmma4(a1, w1f[0], c0);
    c1 = wmma4(a1, w1f[1], c1);
    __builtin_amdgcn_wave_barrier();              // order vs previous-iter reads
#pragma unroll
    for (int v = 0; v < 8; v++) {
      hbuf[wv][(v + 8 * half) * 32 + col] = swishf(c0[v] + b1v0);
      hbuf[wv][(v + 8 * half) * 32 + 16 + col] = swishf(c1[v] + b1v1);
    }
    __builtin_amdgcn_wave_barrier();
    // ---- layer 2: 16x32 @ 32x32 ----
    v8f d0 = {}, d1 = {};
#pragma unroll
    for (int kc = 0; kc < 8; kc++) {
      v2f a2;
      a2.x = hbuf[wv][row * 32 + 4 * kc + 2 * half];
      a2.y = hbuf[wv][row * 32 + 4 * kc + 1 + 2 * half];
      d0 = wmma4(a2, w2f[kc][0], d0);
      d1 = wmma4(a2, w2f[kc][1], d1);
    }
    __builtin_amdgcn_wave_barrier();
#pragma unroll
    for (int v = 0; v < 8; v++) {
      hbuf[wv][(v + 8 * half) * 32 + col] = swishf(d0[v] + b2v0);
      hbuf[wv][(v + 8 * half) * 32 + 16 + col] = swishf(d1[v] + b2v1);
    }
    __builtin_amdgcn_wave_barrier();
    // ---- layer 3: 16x32 @ 32x16 ----
    v8f e0 = {};
#pragma unroll
    for (int kc = 0; kc < 8; kc++) {
      v2f a3;
      a3.x = hbuf[wv][row * 32 + 4 * kc + 2 * half];
      a3.y = hbuf[wv][row * 32 + 4 * kc + 1 + 2 * half];
      e0 = wmma4(a3, w3f[kc], e0);
    }
    float s = 0.f;
#pragma unroll
    for (int v = 0; v < 8; v++) s += swishf(e0[v] + b3v);
    s += __shfl_xor(s, 16, 32);                   // combine M=0..7 and M=8..15
    if (lane < 16) wagg[(p * KPAD + j) * 16 + lane] = s * validf;
  }
  if (lane < 16) {
    for (int jp = KNB; jp < KPAD; jp++) wagg[(p * KPAD + jp) * 16 + lane] = 0.f;
  }
}

// ---------------------------------------------------------------------------
// Kernel 3: one wave per (batch, 16-pixel tile).
// Phase 1: P(32x16) = gather(x)^T @ wagg_p per pixel  -> Pflat[16][512] in LDS
// Phase 2: out(16x64) = Pflat @ Wl, * 1/L, + bl
// ---------------------------------------------------------------------------
__global__ void __launch_bounds__(32)
k_out(const float* __restrict__ x, const int* __restrict__ nbidx,
      const float* __restrict__ wagg, const float* __restrict__ Wl,
      const float* __restrict__ bl, float* __restrict__ out) {
  __shared__ float Pst[16 * 512];
  int lane = threadIdx.x & 31;
  int row = lane & 15;
  int half = lane >> 4;
  int bb = blockIdx.x / 144;
  int p0 = (blockIdx.x % 144) * 16;

  for (int pl = 0; pl < 16; pl++) {
    int p = p0 + pl;
    v8f acc0 = {}, acc1 = {};
#pragma unroll
    for (int kc = 0; kc < 8; kc++) {
      int j0 = 4 * kc + 2 * half;
      int j1 = j0 + 1;
      v2f bfrag;                                  // B = wagg_p (j x f), zero-padded
      bfrag.x = wagg[(p * KPAD + j0) * 16 + row];
      bfrag.y = wagg[(p * KPAD + j1) * 16 + row];
      int pix0 = 0, pix1 = 0;
      if (j0 < KNB) { int t = nbidx[p * KNB + j0]; pix0 = t < 0 ? 0 : t; }
      if (j1 < KNB) { int t = nbidx[p * KNB + j1]; pix1 = t < 0 ? 0 : t; }
      v2f a0, a1v;                                // A rows = channels, gathered x
      a0.x = x[(bb * CIN + row) * NPIX + pix0];
      a0.y = x[(bb * CIN + row) * NPIX + pix1];
      a1v.x = x[(bb * CIN + 16 + row) * NPIX + pix0];
      a1v.y = x[(bb * CIN + 16 + row) * NPIX + pix1];
      acc0 = wmma4(a0, bfrag, acc0);
      acc1 = wmma4(a1v, bfrag, acc1);
    }
#pragma unroll
    for (int v = 0; v < 8; v++) {
      int c0 = v + 8 * half;
      Pst[pl * 512 + c0 * 16 + row] = acc0[v];
      Pst[pl * 512 + (16 + c0) * 16 + row] = acc1[v];
    }
  }
  __builtin_amdgcn_wave_barrier();                // single-wave block
  v8f acc[4] = {{}, {}, {}, {}};
  for (int kc = 0; kc < 128; kc++) {
    v2f a;                                        // A rows = pixels, cols = (c,f)
    a.x = Pst[row * 512 + 4 * kc + 2 * half];
    a.y = Pst[row * 512 + 4 * kc + 1 + 2 * half];
    int k0 = 4 * kc + 2 * half;
#pragma unroll
    for (int t = 0; t < 4; t++) {
      v2f b;
      b.x = Wl[k0 * 64 + t * 16 + row];
      b.y = Wl[(k0 + 1) * 64 + t * 16 + row];
      acc[t] = wmma4(a, b, acc[t]);
    }
  }
#pragma unroll
  for (int t = 0; t < 4; t++) {
#pragma unroll
    for (int v = 0; v < 8; v++) {
      int o = t * 16 + row;
      int pl = v + 8 * half;
      out[(bb * COUT + o) * NPIX + p0 + pl] = acc[t][v] * 0.25f + bl[o];
    }
  }
}

extern "C" void kernel_launch(void* const* d_in, const int* in_sizes, int n_in,
                              void* d_out, int out_size, void* d_ws, size_t ws_size,
                              hipStream_t stream) {
  const float* x  = (const float*)d_in[0];
  const int*   nb = (const int*)d_in[1];
  const float* W1 = (const float*)d_in[2];
  const float* b1 = (const float*)d_in[3];
  const float* W2 = (const float*)d_in[4];
  const float* b2 = (const float*)d_in[5];
  const float* W3 = (const float*)d_in[6];
  const float* b3 = (const float*)d_in[7];
  const float* Wl = (const float*)d_in[8];
  const float* bl = (const float*)d_in[9];
  float* out  = (float*)d_out;
  float* Rbuf = (float*)d_ws;               // 9216*12 floats  (~432 KB)
  float* wagg = Rbuf + NL * RSTRIDE;        // 2304*32*16 floats (~4.7 MB)

  k_build_R<<<(NL + 255) / 256, 256, 0, stream>>>(Rbuf);
  k_wagg<<<NPIX / 4, 128, 0, stream>>>(Rbuf, nb, W1, b1, W2, b2, W3, b3, wagg);
  k_out<<<2 * (NPIX / 16), 32, 0, stream>>>(x, nb, wagg, Wl, bl, out);
}